// AffinityPropagate1_34265249088047
// MI455X (gfx1250) — compile-verified
//
#include <hip/hip_runtime.h>
#include <hip/hip_fp16.h>
#include <stdint.h>

// Problem dimensions (fixed by the reference).
#define BB   8
#define HH   480
#define WW   640
#define HWs  (HH * WW)                 // 307200
#define NPIX (BB * HWs)                // 2457600
#define KELEM ((size_t)BB * 9 * HWs)   // elements per softmaxed guided tensor

// Tile geometry: each 256-thread block computes 128x8 pixels (4 px / thread).
#define TX   128
#define TY   8
#define HALO 3                          // max dilation
#define LW   (TX + 2 * HALO)            // 134
#define LH   (TY + 2 * HALO)            // 14
#define LWP  136                        // padded LDS row stride (floats)

typedef float     f4 __attribute__((ext_vector_type(4)));
typedef _Float16  h4 __attribute__((ext_vector_type(4)));

// ---------------------------------------------------------------------------
// Async global -> LDS staging (CDNA5 ASYNCcnt path), with fallbacks.
// ---------------------------------------------------------------------------
typedef __attribute__((address_space(1))) int g1int;
typedef __attribute__((address_space(3))) int l3int;

__device__ __forceinline__ void stage_b32(float* lds_dst, const float* gsrc) {
#if defined(__gfx1250__) && __has_builtin(__builtin_amdgcn_global_load_async_to_lds_b32)
  // GLOBAL_LOAD_ASYNC_TO_LDS_B32: per-lane 4B copy, tracked by ASYNCcnt.
  __builtin_amdgcn_global_load_async_to_lds_b32(
      (g1int*)(uintptr_t)gsrc,
      (l3int*)(uint32_t)(uintptr_t)lds_dst,
      /*offset=*/0, /*cpol=*/0);
#elif __has_builtin(__builtin_amdgcn_load_to_lds)
  __builtin_amdgcn_load_to_lds((g1int*)(uintptr_t)gsrc,
                               (l3int*)(uint32_t)(uintptr_t)lds_dst,
                               /*size=*/4, /*offset=*/0, /*aux=*/0);
#else
  *lds_dst = *gsrc;
#endif
}

__device__ __forceinline__ void wait_async_stage() {
#if defined(__gfx1250__) && __has_builtin(__builtin_amdgcn_s_wait_asynccnt)
  __builtin_amdgcn_s_wait_asynccnt(0);
#elif defined(__gfx1250__)
  asm volatile("s_wait_asynccnt 0" ::: "memory");
#endif
}

// ---------------------------------------------------------------------------
// Kernel 1: softmax over the 9-channel axis of guided1/2/3, stored as f16.
// f16 weights (132.7 MB) keep the whole per-iteration working set inside the
// MI455X 192 MB L2, so 7 of 8 propagate passes never touch HBM. The guided
// inputs are read exactly once -> non-temporal loads so this 265 MB one-shot
// stream does not evict the weights we want L2-resident.
// 4 pixels per thread: b128 NT loads, b64 stores.
// ---------------------------------------------------------------------------
__global__ __launch_bounds__(256)
void softmax9_f16(const float* __restrict__ g0, const float* __restrict__ g1,
                  const float* __restrict__ g2, __half* __restrict__ kout) {
  const int g = blockIdx.y;                              // which guided tensor
  const int p = (blockIdx.x * 256 + threadIdx.x) * 4;    // pixel quad
  const float* src = (g == 0) ? g0 : ((g == 1) ? g1 : g2);
  __half* dst = kout + (size_t)g * KELEM;

  const int b = p / HWs;
  const int s = p - b * HWs;
  const size_t base = (size_t)b * 9 * HWs + s;

  f4 v[9];
  f4 m = {-INFINITY, -INFINITY, -INFINITY, -INFINITY};
#pragma unroll
  for (int c = 0; c < 9; ++c) {
    v[c] = __builtin_nontemporal_load((const f4*)(src + base + (size_t)c * HWs));
#pragma unroll
    for (int q = 0; q < 4; ++q) m[q] = fmaxf(m[q], v[c][q]);
  }
  f4 sum = {0.f, 0.f, 0.f, 0.f};
#pragma unroll
  for (int c = 0; c < 9; ++c) {
#pragma unroll
    for (int q = 0; q < 4; ++q) {
      v[c][q] = __expf(v[c][q] - m[q]);
      sum[q] += v[c][q];
    }
  }
  f4 r;
#pragma unroll
  for (int q = 0; q < 4; ++q) r[q] = 1.0f / sum[q];
#pragma unroll
  for (int c = 0; c < 9; ++c) {
    h4 o;
#pragma unroll
    for (int q = 0; q < 4; ++q) o[q] = (_Float16)(v[c][q] * r[q]);
    *(h4*)(dst + base + (size_t)c * HWs) = o;            // b64, L2-resident
  }
}

// ---------------------------------------------------------------------------
// Kernel 2: one propagation iteration, 4 horizontally adjacent pixels/thread.
//   out(p) = f1*sum(N1(x)*k1) + f2*sum(N2(x)*k2) + f3*sum(N3(x)*k3)
// x tile (+halo 3) staged into LDS with async loads. Interior blocks skip
// the zero-fill (every used LDS element is overwritten); boundary blocks
// zero-fill first to reproduce the reference's zero padding.
// ---------------------------------------------------------------------------
__global__ __launch_bounds__(256)
void propagate(const float* __restrict__ xin, const __half* __restrict__ kall,
               const float* __restrict__ fuse, float* __restrict__ xout) {
  __shared__ float tile[LH][LWP];

  const int tid = threadIdx.x;
  const int tx = tid & 31;
  const int ty = tid >> 5;                 // one wave32 per tile row
  const int x0 = blockIdx.x * TX;
  const int y0 = blockIdx.y * TY;
  const int b  = blockIdx.z;

  const float* xplane = xin + (size_t)b * HWs;

  const bool interior = (x0 >= HALO) && (y0 >= HALO) &&
                        (x0 + TX + HALO <= WW) && (y0 + TY + HALO <= HH);

  if (interior) {
    // Every element of the 134x14 used region is written by one async load.
#pragma unroll
    for (int r = 0; r < 8; ++r) {          // ceil(134*14 / 256) = 8 rounds
      int i = r * 256 + tid;
      if (i < LH * LW) {
        int ly = i / LW;
        int lx = i - ly * LW;
        stage_b32(&tile[ly][lx],
                  xplane + (y0 - HALO + ly) * WW + (x0 - HALO + lx));
      }
    }
  } else {
    // Zero-fill (border padding), then stage only the in-bounds elements.
    for (int i = tid; i < LH * LWP; i += 256) (&tile[0][0])[i] = 0.f;
    __syncthreads();
#pragma unroll
    for (int r = 0; r < 8; ++r) {
      int i = r * 256 + tid;
      if (i < LH * LW) {
        int ly = i / LW;
        int lx = i - ly * LW;
        int gy = y0 - HALO + ly;
        int gx = x0 - HALO + lx;
        if (gy >= 0 && gy < HH && gx >= 0 && gx < WW)
          stage_b32(&tile[ly][lx], xplane + gy * WW + gx);
      }
    }
  }
  wait_async_stage();
  __syncthreads();

  const int gx = x0 + 4 * tx;              // multiple of 4 -> b64/b128 aligned
  const int gy = y0 + ty;
  const int s = gy * WW + gx;
  const size_t kb = (size_t)b * 9 * HWs + s;
  const __half* k1 = kall;
  const __half* k2 = kall + KELEM;
  const __half* k3 = kall + 2 * KELEM;

  const int ly = ty + HALO, lx = 4 * tx + HALO;
  float o1[4] = {0.f, 0.f, 0.f, 0.f};
  float o2[4] = {0.f, 0.f, 0.f, 0.f};
  float o3[4] = {0.f, 0.f, 0.f, 0.f};
#pragma unroll
  for (int i = 0; i < 3; ++i) {
#pragma unroll
    for (int j = 0; j < 3; ++j) {
      const int c = i * 3 + j;
      const size_t ko = kb + (size_t)c * HWs;
      const h4 w1 = *(const h4*)(k1 + ko);      // b64: 4 pixels' weights
      const h4 w2 = *(const h4*)(k2 + ko);
      const h4 w3 = *(const h4*)(k3 + ko);
      const int r1 = ly + (i - 1),     c1 = lx + (j - 1);
      const int r2 = ly + 2 * (i - 1), c2 = lx + 2 * (j - 1);
      const int r3 = ly + 3 * (i - 1), c3 = lx + 3 * (j - 1);
#pragma unroll
      for (int q = 0; q < 4; ++q) {
        o1[q] = fmaf(tile[r1][c1 + q], (float)w1[q], o1[q]);
        o2[q] = fmaf(tile[r2][c2 + q], (float)w2[q], o2[q]);
        o3[q] = fmaf(tile[r3][c3 + q], (float)w3[q], o3[q]);
      }
    }
  }

  const size_t fb = (size_t)b * 3 * HWs + s;
  const f4 f1 = *(const f4*)(fuse + fb);        // b128
  const f4 f2 = *(const f4*)(fuse + fb + HWs);
  const f4 f3 = *(const f4*)(fuse + fb + 2 * (size_t)HWs);

  f4 res;
#pragma unroll
  for (int q = 0; q < 4; ++q)
    res[q] = o1[q] * f1[q] + o2[q] * f2[q] + o3[q] * f3[q];
  *(f4*)(xout + (size_t)b * HWs + s) = res;     // b128
}

// ---------------------------------------------------------------------------
// Launch: softmax once, then 8 propagate passes ping-ponging ws <-> d_out
// (even count => final result lands in d_out). Deterministic, stream-ordered.
// ---------------------------------------------------------------------------
extern "C" void kernel_launch(void* const* d_in, const int* in_sizes, int n_in,
                              void* d_out, int out_size, void* d_ws, size_t ws_size,
                              hipStream_t stream) {
  const float* guided1 = (const float*)d_in[0];
  const float* guided2 = (const float*)d_in[1];
  const float* guided3 = (const float*)d_in[2];
  const float* fuse    = (const float*)d_in[3];
  const float* x       = (const float*)d_in[4];
  float* out = (float*)d_out;

  // Workspace layout: [3 * KELEM f16 softmax weights][NPIX f32 ping buffer]
  __half* kws  = (__half*)d_ws;
  float*  xtmp = (float*)((char*)d_ws + 3 * KELEM * sizeof(__half));

  {
    dim3 sgrid(NPIX / 1024, 3);                    // 2400 x 3 blocks
    softmax9_f16<<<sgrid, 256, 0, stream>>>(guided1, guided2, guided3, kws);
  }

  dim3 grid(WW / TX, HH / TY, BB);                 // 5 x 60 x 8
  const float* cur = x;
  for (int it = 0; it < 8; ++it) {
    float* nxt = (it & 1) ? out : xtmp;            // it=7 -> out
    propagate<<<grid, 256, 0, stream>>>(cur, kws, fuse, nxt);
    cur = nxt;
  }
}